// hpNet_rela_bi_pur_77223511982422
// MI455X (gfx1250) — compile-verified
//
#include <hip/hip_runtime.h>
#include <hip/hip_bf16.h>
#include <math.h>

typedef float v2f __attribute__((ext_vector_type(2)));
typedef float v8f __attribute__((ext_vector_type(8)));

#define MIN_NORM_F 1e-15f
#define PROJ_EPS_F 4e-3f
#define MAXNORM_F (1.0f - PROJ_EPS_F)
#define ATANH_EPS_F 1e-7f

__device__ __forceinline__ float wave_allsum(float v) {
    for (int o = 16; o >= 1; o >>= 1) v += __shfl_xor(v, o, 32);
    return v;
}
// combine the two half-wave partial sums from the WMMA C/D replicated layout
__device__ __forceinline__ float half_combine(float local) {
    return local + __shfl_xor(local, 16, 32);
}
__device__ __forceinline__ float artanh_f(float x) {
    x = fminf(fmaxf(x, -1.0f + ATANH_EPS_F), 1.0f - ATANH_EPS_F);
    return 0.5f * (log1pf(x) - log1pf(-x));
}
// scale s so that proj(expmap0(u)) == s*u ; also returns ||s*u||^2
__device__ __forceinline__ float ptan0exp_scale(float n2, float* vsq) {
    float un = sqrtf(fmaxf(n2, MIN_NORM_F));
    float s  = tanhf(un) / un;
    float vn = sqrtf(fmaxf(s * s * n2, MIN_NORM_F));
    if (vn > MAXNORM_F) s *= MAXNORM_F / vn;
    *vsq = s * s * n2;
    return s;
}
// sqdist(p1,p2) with x=-p1: caller passes x2=||p1||^2, y2=||p2||^2, xy=<-p1,p2>
__device__ __forceinline__ float sqdist_s(float x2, float y2, float xy) {
    float alpha = 1.0f + 2.0f * xy + y2;
    float beta  = 1.0f - x2;
    float num2  = alpha * alpha * x2 + 2.0f * alpha * beta * xy + beta * beta * y2;
    float denom = fmaxf(1.0f + 2.0f * xy + x2 * y2, MIN_NORM_F);
    float sq    = num2 / (denom * denom);
    float dn    = sqrtf(fmaxf(sq, MIN_NORM_F));
    float d     = 2.0f * artanh_f(dn);
    return d * d;
}
__device__ __forceinline__ v8f wmma4(v2f a, v2f b, v8f c) {
    return __builtin_amdgcn_wmma_f32_16x16x4_f32(false, a, false, b, (short)0, c,
                                                 false, false);
}
__device__ __forceinline__ v8f vzero8() {
    v8f z; for (int i = 0; i < 8; ++i) z[i] = 0.0f; return z;
}

// ---------------------------------------------------------------------------
// Kernel 1: per 16 items (1 wave): Ti = pre_diag @ item  via WMMA f32 16x16x4,
// then item_ri = proj_tan0_exp(Ti) and Ri = proj_tan0_exp(item) to workspace.
// ---------------------------------------------------------------------------
__global__ __launch_bounds__(32) void k1_item(const int* __restrict__ items,
                                              const float* __restrict__ ent,
                                              const float* __restrict__ rel,
                                              float* __restrict__ wsRi,
                                              float* __restrict__ wsItemRi,
                                              int B) {
    __shared__ float lit[16 * 64];
    const int l = threadIdx.x;
    const int base = blockIdx.x * 16;
    for (int idx = l; idx < 16 * 64; idx += 32) {
        int it = items[base + (idx >> 6)];
        lit[idx] = ent[it * 64 + (idx & 63)];
    }
    __syncthreads();

    const float* pd = rel + 32 * 64 * 64;   // relation_emb[N_RELATION]
    const int row  = l & 15;
    const int koff = (l < 16) ? 0 : 2;
    const int hb   = (l >= 16) ? 8 : 0;

    v8f acc[4];
    for (int t = 0; t < 4; ++t) acc[t] = vzero8();
    for (int s = 0; s < 16; ++s) {
        int k0 = 4 * s;
        v2f b;  // column `row`'s item vector, B 4x16 tile
        b.x = lit[row * 64 + k0 + koff];
        b.y = lit[row * 64 + k0 + koff + 1];
        for (int t = 0; t < 4; ++t) {
            const float* ap = pd + (t * 16 + row) * 64 + k0 + koff;
            v2f a; a.x = ap[0]; a.y = ap[1];
            acc[t] = wmma4(a, b, acc[t]);
        }
    }
    // lane l holds comps {16t + hb + v} of Ti for item column (l&15)
    float n2loc = 0.0f;
    for (int t = 0; t < 4; ++t)
        for (int v = 0; v < 8; ++v) n2loc += acc[t][v] * acc[t][v];
    float n2 = half_combine(n2loc);
    float hsq;
    float s1 = ptan0exp_scale(n2, &hsq);
    const int gb = base + row;
    for (int t = 0; t < 4; ++t) {
        float* o = wsItemRi + gb * 64 + 16 * t + hb;
        float4 a0 = make_float4(s1*acc[t][0], s1*acc[t][1], s1*acc[t][2], s1*acc[t][3]);
        float4 a1 = make_float4(s1*acc[t][4], s1*acc[t][5], s1*acc[t][6], s1*acc[t][7]);
        ((float4*)o)[0] = a0;
        ((float4*)o)[1] = a1;
    }
    // Ri = proj_tan0_exp(item_emb) in the same replicated-half layout
    float rv[32];
    float m2loc = 0.0f;
    for (int t = 0; t < 4; ++t)
        for (int v = 0; v < 8; ++v) {
            float x = lit[row * 64 + 16 * t + hb + v];
            rv[t * 8 + v] = x;
            m2loc += x * x;
        }
    float m2 = half_combine(m2loc);
    float rsq;
    float s2 = ptan0exp_scale(m2, &rsq);
    for (int t = 0; t < 4; ++t) {
        float* o = wsRi + gb * 64 + 16 * t + hb;
        float4 a0 = make_float4(s2*rv[t*8+0], s2*rv[t*8+1], s2*rv[t*8+2], s2*rv[t*8+3]);
        float4 a1 = make_float4(s2*rv[t*8+4], s2*rv[t*8+5], s2*rv[t*8+6], s2*rv[t*8+7]);
        ((float4*)o)[0] = a0;
        ((float4*)o)[1] = a1;
    }
}

// ---------------------------------------------------------------------------
// Kernel 2: one wave per (h,b). For each of M memory slots: WMMA matvec
// Rh = (R[r]/10) @ h_emb (h-vector replicated across the 16 B-columns),
// attention prob via analytic mobius/sqdist scalars, Klein t-path, then
// softmax-normalized weighted sum -> kle_o[b,h,:].
// ---------------------------------------------------------------------------
__global__ __launch_bounds__(32) void k2_attn(const int* __restrict__ mh,
                                              const int* __restrict__ mr,
                                              const int* __restrict__ mt,
                                              const float* __restrict__ ent,
                                              const float* __restrict__ rel,
                                              const float* __restrict__ wsRi,
                                              float* __restrict__ wsKleo,
                                              int B, int M) {
    __shared__ float kleS[32 * 64];
    __shared__ float probS[32];
    __shared__ float wS[32];
    const int l = threadIdx.x;
    const int bid = blockIdx.x;       // 0 .. 2B-1
    const int h = bid / B;
    const int b = bid - h * B;
    const int row  = l & 15;
    const int koff = (l < 16) ? 0 : 2;
    const int hb   = (l >= 16) ? 8 : 0;

    // Ri[b] in replicated-half layout + its squared norm
    float riv[32];
    float ri2loc = 0.0f;
    for (int t = 0; t < 4; ++t) {
        const float4* p = (const float4*)(wsRi + b * 64 + 16 * t + hb);
        float4 u0 = p[0], u1 = p[1];
        riv[t*8+0]=u0.x; riv[t*8+1]=u0.y; riv[t*8+2]=u0.z; riv[t*8+3]=u0.w;
        riv[t*8+4]=u1.x; riv[t*8+5]=u1.y; riv[t*8+6]=u1.z; riv[t*8+7]=u1.w;
        ri2loc += u0.x*u0.x + u0.y*u0.y + u0.z*u0.z + u0.w*u0.w;
        ri2loc += u1.x*u1.x + u1.y*u1.y + u1.z*u1.z + u1.w*u1.w;
    }
    const float ri2 = half_combine(ri2loc);

    for (int m = 0; m < M; ++m) {
        const int idx = (h * B + b) * M + m;
        const int hE = mh[idx];
        const int r  = mr[idx];
        const int tE = mt[idx];

        const float* R = rel + r * 4096;
        __builtin_prefetch(R, 0, 0);

        // h_emb/10 as float2 per lane (comps 2l, 2l+1)
        float2 xv = ((const float2*)(ent + hE * 64))[l];
        xv.x *= 0.1f; xv.y *= 0.1f;

        v8f acc[4];
        for (int t = 0; t < 4; ++t) acc[t] = vzero8();
        for (int s = 0; s < 16; ++s) {
            const int k0 = 4 * s;
            const int src = 2 * s + ((l >= 16) ? 1 : 0);
            v2f bb;
            bb.x = __shfl(xv.x, src, 32);
            bb.y = __shfl(xv.y, src, 32);
            for (int t = 0; t < 4; ++t) {
                const float* ap = R + (t * 16 + row) * 64 + k0 + koff;
                v2f a; a.x = ap[0]; a.y = ap[1];
                acc[t] = wmma4(a, bb, acc[t]);
            }
        }
        // reductions: ||Rh||^2 and <Rh, Ri>
        float n2loc = 0.0f, dloc = 0.0f;
        for (int t = 0; t < 4; ++t)
            for (int v = 0; v < 8; ++v) {
                float yv = acc[t][v];
                n2loc += yv * yv;
                dloc  += yv * riv[t * 8 + v];
            }
        float n2   = half_combine(n2loc);
        float dotd = half_combine(dloc);
        float hx2;
        float s1 = ptan0exp_scale(n2, &hx2);       // Rh_hyp = s1*Rh, ||.||^2=hx2
        float xy = -s1 * dotd;                     // <-Rh_hyp, Ri_hyp>
        float d2 = sqdist_s(hx2, ri2, xy);
        float prob = expf(-0.2f * d2 + 0.05f);

        // t path: kle_t = sk * t_emb (simple float2/lane layout)
        float2 tv = ((const float2*)(ent + tE * 64))[l];
        float t2 = wave_allsum(tv.x * tv.x + tv.y * tv.y);
        float th2;
        float se = ptan0exp_scale(t2, &th2);
        float sk = se * (2.0f / (1.0f + th2));     // expmap0+proj then to_klein
        float kle2 = sk * sk * t2;
        float lf = rsqrtf(fmaxf(1.0f - kle2, MIN_NORM_F));
        prob *= lf;

        ((float2*)kleS)[m * 32 + l] = make_float2(sk * tv.x, sk * tv.y);
        if (l == 0) probS[m] = prob;
    }
    __syncthreads();
    float pm = (l < M) ? probS[l] : 0.0f;
    float psum = wave_allsum(pm);
    if (l < M) wS[l] = pm / psum;
    __syncthreads();
    float ax = 0.0f, ay = 0.0f;
    for (int m = 0; m < M; ++m) {
        float w = wS[m];
        float2 k = ((const float2*)kleS)[m * 32 + l];
        ax += k.x * w;
        ay += k.y * w;
    }
    ((float2*)(wsKleo + (b * 2 + h) * 64))[l] = make_float2(ax, ay);
}

// ---------------------------------------------------------------------------
// Kernel 3: one wave per batch element: lorentz-weighted hop combine,
// klein_to + proj, sqdist vs item_ri, fermi-dirac score.
// ---------------------------------------------------------------------------
__global__ __launch_bounds__(32) void k3_final(const float* __restrict__ wsKleo,
                                               const float* __restrict__ wsItemRi,
                                               float* __restrict__ out, int B) {
    const int b = blockIdx.x;
    const int l = threadIdx.x;
    float2 k0 = ((const float2*)(wsKleo + (b * 2 + 0) * 64))[l];
    float2 k1 = ((const float2*)(wsKleo + (b * 2 + 1) * 64))[l];
    float q0 = wave_allsum(k0.x * k0.x + k0.y * k0.y);
    float q1 = wave_allsum(k1.x * k1.x + k1.y * k1.y);
    float g0 = rsqrtf(fmaxf(1.0f - q0, MIN_NORM_F));
    float g1 = rsqrtf(fmaxf(1.0f - q1, MIN_NORM_F));
    float inv = 1.0f / (g0 + g1);
    float w0 = g0 * inv, w1 = g1 * inv;
    float fx = w0 * k0.x + w1 * k1.x;
    float fy = w0 * k0.y + w1 * k1.y;
    float kf2 = wave_allsum(fx * fx + fy * fy);
    float kt = 1.0f / (1.0f + sqrtf(fmaxf(1.0f - kf2, MIN_NORM_F)));
    float hx = kt * fx, hy = kt * fy;
    float h2 = kt * kt * kf2;
    float hn = sqrtf(fmaxf(h2, MIN_NORM_F));
    if (hn > MAXNORM_F) {
        float sp = MAXNORM_F / hn;
        hx *= sp; hy *= sp; h2 = sp * sp * h2;
    }
    float2 ir = ((const float2*)(wsItemRi + b * 64))[l];
    float y2  = wave_allsum(ir.x * ir.x + ir.y * ir.y);
    float dot = wave_allsum(hx * ir.x + hy * ir.y);
    float d2 = sqdist_s(h2, y2, -dot);
    float score = 1.0f / (expf(d2 - 2.0f) + 1.0f);
    if (l == 0) out[b] = score;
}

extern "C" void kernel_launch(void* const* d_in, const int* in_sizes, int n_in,
                              void* d_out, int out_size, void* d_ws, size_t ws_size,
                              hipStream_t stream) {
    const int*   items = (const int*)d_in[1];
    const int*   mh    = (const int*)d_in[3];
    const int*   mr    = (const int*)d_in[4];
    const int*   mt    = (const int*)d_in[5];
    const float* ent   = (const float*)d_in[6];
    const float* rel   = (const float*)d_in[7];

    const int B = in_sizes[1];               // 512
    const int M = in_sizes[3] / (2 * B);     // 32 (H = 2)

    float* ws       = (float*)d_ws;
    float* wsRi     = ws;                    // [B,64]
    float* wsItemRi = ws + (size_t)B * 64;   // [B,64]
    float* wsKleo   = ws + (size_t)2 * B * 64; // [B,2,64]

    k1_item<<<B / 16, 32, 0, stream>>>(items, ent, rel, wsRi, wsItemRi, B);
    k2_attn<<<2 * B, 32, 0, stream>>>(mh, mr, mt, ent, rel, wsRi, wsKleo, B, M);
    k3_final<<<B, 32, 0, stream>>>(wsKleo, wsItemRi, (float*)d_out, B);
}